// MPNNDiff_16484084483096
// MI455X (gfx1250) — compile-verified
//
#include <hip/hip_runtime.h>

// MPNN forward for MI455X (gfx1250), exact-f32 math via V_WMMA_F32_16X16X4_F32.
// d_ws layout: summed[N*64] | counts[N] | W1pad[132*128] (f32 each).

typedef __attribute__((ext_vector_type(2))) float v2f;
typedef __attribute__((ext_vector_type(8))) float v8f;

#define WAVES 4

// ---------------------------------------------------------------------------
// init: zero summed+counts, build zero-padded W1 (131x128 -> 132x128)
// ---------------------------------------------------------------------------
__global__ __launch_bounds__(256) void init_ws_kernel(float* __restrict__ ws,
                                                      const float* __restrict__ W1,
                                                      int N) {
  long i = (long)blockIdx.x * 256 + threadIdx.x;
  long nsum = (long)N * 64 + N;            // summed + counts
  long tot  = nsum + 132 * 128;            // + W1pad
  if (i < nsum) {
    ws[i] = 0.0f;
  } else if (i < tot) {
    long j = i - nsum;                     // j = k*128 + c
    int k = (int)(j >> 7);
    ws[i] = (k < 131) ? W1[j] : 0.0f;
  }
}

// ---------------------------------------------------------------------------
// edge kernel: one wave handles 16 edges.
//   msg_in = [x_i | x_j - x_i | rel_pos | 0pad]  (16 x 132, LDS)
//   hmsg   = relu(msg_in @ W1 + b1)              (16 x 128, VGPR acc -> LDS)
//   msg    = hmsg @ W2 + b2                      (16 x 64,  VGPR acc)
//   summed[idx_i] += msg ; counts[idx_i] += 1    (global f32 atomics)
// ---------------------------------------------------------------------------
__global__ __launch_bounds__(32 * WAVES) void edge_kernel(
    const float* __restrict__ x, const float* __restrict__ pos,
    const int* __restrict__ ei, const float* __restrict__ W1p,
    const float* __restrict__ b1, const float* __restrict__ W2,
    const float* __restrict__ b2, float* __restrict__ summed,
    float* __restrict__ counts, int N, int E) {
  __shared__ float smem[WAVES][16 * 132];
  const int lane = threadIdx.x & 31;
  const int wave = threadIdx.x >> 5;
  float* T = smem[wave];

  const long t  = (long)blockIdx.x * WAVES + wave;
  const long e0 = t * 16;
  const int* idx_i = ei;        // edge_index[0][:]
  const int* idx_j = ei + E;    // edge_index[1][:]

  const int mrow  = lane & 15;          // M (or N) position within tile
  const int khalf = (lane >> 4) << 1;   // K sub-offset per WMMA f32 layout
  const int mhi   = (lane >> 4) << 3;   // C/D: lanes>=16 hold rows M+8

  // ---- stage msg_in tile ----
  {
    const long e  = e0 + mrow;
    const long ec = (e < E) ? e : (long)(E - 1);
    const int ni = idx_i[ec];
    const int nj = idx_j[ec];
    if (lane < 16) {
      const float4* xi4 = (const float4*)(x + (long)ni * 64);
      float4* dst = (float4*)(T + mrow * 132);
      #pragma unroll
      for (int c = 0; c < 16; ++c) dst[c] = xi4[c];
    } else {
      const float4* xi4 = (const float4*)(x + (long)ni * 64);
      const float4* xj4 = (const float4*)(x + (long)nj * 64);
      float4* dst = (float4*)(T + mrow * 132 + 64);
      #pragma unroll
      for (int c = 0; c < 16; ++c) {
        float4 a = xi4[c], b = xj4[c];
        dst[c] = make_float4(b.x - a.x, b.y - a.y, b.z - a.z, b.w - a.w);
      }
      float* p = T + mrow * 132 + 128;
      p[0] = pos[(long)nj * 3 + 0] - pos[(long)ni * 3 + 0];
      p[1] = pos[(long)nj * 3 + 1] - pos[(long)ni * 3 + 1];
      p[2] = pos[(long)nj * 3 + 2] - pos[(long)ni * 3 + 2];
      p[3] = 0.0f;  // K=131 pad
    }
  }
  __syncthreads();

  // ---- MLP1: 16x132 @ 132x128 ----
  v8f acc[8] = {};
  for (int kk = 0; kk < 132; kk += 4) {
    const int k0 = kk + khalf;
    v2f a;
    a.x = T[mrow * 132 + k0];
    a.y = T[mrow * 132 + k0 + 1];
    #pragma unroll
    for (int nt = 0; nt < 8; ++nt) {
      const int col = nt * 16 + mrow;
      v2f b;
      b.x = W1p[k0 * 128 + col];
      b.y = W1p[(k0 + 1) * 128 + col];
      acc[nt] = __builtin_amdgcn_wmma_f32_16x16x4_f32(
          false, a, false, b, (short)0, acc[nt], false, false);
    }
  }
  __syncthreads();

  // ---- relu + b1, stage hmsg as 16x128 over the same LDS buffer ----
  #pragma unroll
  for (int nt = 0; nt < 8; ++nt) {
    const int col = nt * 16 + mrow;
    const float bb = b1[col];
    #pragma unroll
    for (int r = 0; r < 8; ++r) {
      float v = acc[nt][r] + bb;
      T[(mhi + r) * 128 + col] = (v > 0.0f) ? v : 0.0f;
    }
  }
  __syncthreads();

  // ---- MLP2: 16x128 @ 128x64 ----
  v8f acc2[4] = {};
  for (int kk = 0; kk < 128; kk += 4) {
    const int k0 = kk + khalf;
    v2f a;
    a.x = T[mrow * 128 + k0];
    a.y = T[mrow * 128 + k0 + 1];
    #pragma unroll
    for (int nt = 0; nt < 4; ++nt) {
      const int col = nt * 16 + mrow;
      v2f b;
      b.x = W2[k0 * 64 + col];
      b.y = W2[(k0 + 1) * 64 + col];
      acc2[nt] = __builtin_amdgcn_wmma_f32_16x16x4_f32(
          false, a, false, b, (short)0, acc2[nt], false, false);
    }
  }

  // ---- scatter-add into summed / counts ----
  {
    int node[8];
    bool valid[8];
    #pragma unroll
    for (int r = 0; r < 8; ++r) {
      long e = e0 + mhi + r;
      valid[r] = (e < E);
      node[r] = idx_i[valid[r] ? e : (long)(E - 1)];
    }
    #pragma unroll
    for (int nt = 0; nt < 4; ++nt) {
      const int col = nt * 16 + mrow;
      const float bb = b2[col];
      #pragma unroll
      for (int r = 0; r < 8; ++r) {
        if (valid[r])
          atomicAdd(summed + (long)node[r] * 64 + col, acc2[nt][r] + bb);
      }
    }
    if (lane < 16) {
      long e = e0 + lane;
      if (e < E) atomicAdd(counts + idx_i[e], 1.0f);
    }
  }
}

// ---------------------------------------------------------------------------
// node kernel: one wave handles 16 nodes.
//   upd_in = [x | summed/max(counts,1)]  (16 x 128, LDS)
//   hupd   = relu(upd_in @ W3 + b3)      (16 x 128)
//   out    = hupd @ W4 + b4              (16 x 64)
// ---------------------------------------------------------------------------
__global__ __launch_bounds__(32 * WAVES) void node_kernel(
    const float* __restrict__ x, const float* __restrict__ summed,
    const float* __restrict__ counts, const float* __restrict__ W3,
    const float* __restrict__ b3, const float* __restrict__ W4,
    const float* __restrict__ b4, float* __restrict__ out, int N) {
  __shared__ float smem[WAVES][16 * 128];
  const int lane = threadIdx.x & 31;
  const int wave = threadIdx.x >> 5;
  float* T = smem[wave];

  const long t  = (long)blockIdx.x * WAVES + wave;
  const long n0 = t * 16;

  const int mrow  = lane & 15;
  const int khalf = (lane >> 4) << 1;
  const int mhi   = (lane >> 4) << 3;

  // ---- stage upd_in tile ----
  {
    const long n  = n0 + mrow;
    const long nc = (n < N) ? n : (long)(N - 1);
    if (lane < 16) {
      const float4* xs = (const float4*)(x + nc * 64);
      float4* dst = (float4*)(T + mrow * 128);
      #pragma unroll
      for (int c = 0; c < 16; ++c) dst[c] = xs[c];
    } else {
      float cnt = counts[nc];
      float inv = 1.0f / (cnt > 1.0f ? cnt : 1.0f);
      const float4* ss = (const float4*)(summed + nc * 64);
      float4* dst = (float4*)(T + mrow * 128 + 64);
      #pragma unroll
      for (int c = 0; c < 16; ++c) {
        float4 s = ss[c];
        dst[c] = make_float4(s.x * inv, s.y * inv, s.z * inv, s.w * inv);
      }
    }
  }
  __syncthreads();

  // ---- MLP3: 16x128 @ 128x128 ----
  v8f acc[8] = {};
  for (int kk = 0; kk < 128; kk += 4) {
    const int k0 = kk + khalf;
    v2f a;
    a.x = T[mrow * 128 + k0];
    a.y = T[mrow * 128 + k0 + 1];
    #pragma unroll
    for (int nt = 0; nt < 8; ++nt) {
      const int col = nt * 16 + mrow;
      v2f b;
      b.x = W3[k0 * 128 + col];
      b.y = W3[(k0 + 1) * 128 + col];
      acc[nt] = __builtin_amdgcn_wmma_f32_16x16x4_f32(
          false, a, false, b, (short)0, acc[nt], false, false);
    }
  }
  __syncthreads();

  // ---- relu + b3, stage hupd ----
  #pragma unroll
  for (int nt = 0; nt < 8; ++nt) {
    const int col = nt * 16 + mrow;
    const float bb = b3[col];
    #pragma unroll
    for (int r = 0; r < 8; ++r) {
      float v = acc[nt][r] + bb;
      T[(mhi + r) * 128 + col] = (v > 0.0f) ? v : 0.0f;
    }
  }
  __syncthreads();

  // ---- MLP4: 16x128 @ 128x64 ----
  v8f acc2[4] = {};
  for (int kk = 0; kk < 128; kk += 4) {
    const int k0 = kk + khalf;
    v2f a;
    a.x = T[mrow * 128 + k0];
    a.y = T[mrow * 128 + k0 + 1];
    #pragma unroll
    for (int nt = 0; nt < 4; ++nt) {
      const int col = nt * 16 + mrow;
      v2f b;
      b.x = W4[k0 * 64 + col];
      b.y = W4[(k0 + 1) * 64 + col];
      acc2[nt] = __builtin_amdgcn_wmma_f32_16x16x4_f32(
          false, a, false, b, (short)0, acc2[nt], false, false);
    }
  }

  // ---- write out ----
  #pragma unroll
  for (int nt = 0; nt < 4; ++nt) {
    const int col = nt * 16 + mrow;
    const float bb = b4[col];
    #pragma unroll
    for (int r = 0; r < 8; ++r) {
      long n = n0 + mhi + r;
      if (n < N) out[n * 64 + col] = acc2[nt][r] + bb;
    }
  }
}

// ---------------------------------------------------------------------------
extern "C" void kernel_launch(void* const* d_in, const int* in_sizes, int n_in,
                              void* d_out, int out_size, void* d_ws, size_t ws_size,
                              hipStream_t stream) {
  const float* x   = (const float*)d_in[0];
  const float* pos = (const float*)d_in[1];
  const int*   ei  = (const int*)d_in[2];
  const float* W1  = (const float*)d_in[3];
  const float* b1  = (const float*)d_in[4];
  const float* W2  = (const float*)d_in[5];
  const float* b2  = (const float*)d_in[6];
  const float* W3  = (const float*)d_in[7];
  const float* b3  = (const float*)d_in[8];
  const float* W4  = (const float*)d_in[9];
  const float* b4  = (const float*)d_in[10];
  float* out = (float*)d_out;

  const int N = in_sizes[0] / 64;   // x is [N, 64]
  const int E = in_sizes[2] / 2;    // edge_index is [2, E]

  float* summed = (float*)d_ws;
  float* counts = summed + (size_t)N * 64;
  float* W1p    = counts + N;       // 132 x 128 zero-padded W1

  long tot = (long)N * 64 + N + 132 * 128;
  init_ws_kernel<<<(unsigned)((tot + 255) / 256), 256, 0, stream>>>(summed, W1, N);

  long tilesE = ((long)E + 15) / 16;
  edge_kernel<<<(unsigned)((tilesE + WAVES - 1) / WAVES), 32 * WAVES, 0, stream>>>(
      x, pos, ei, W1p, b1, W2, b2, summed, counts, N, E);

  long tilesN = ((long)N + 15) / 16;
  node_kernel<<<(unsigned)((tilesN + WAVES - 1) / WAVES), 32 * WAVES, 0, stream>>>(
      x, summed, counts, W3, b3, W4, b4, out, N);
}